// FPMC_14199161881186
// MI455X (gfx1250) — compile-verified
//
#include <hip/hip_runtime.h>

typedef float v2f __attribute__((ext_vector_type(2)));
typedef float v8f __attribute__((ext_vector_type(8)));

#define N_ITEMS 100000
#define K_IL    64
#define M_ROWS  1024

// C[b,i] = (1/8) * sum_k LI[X[b,8,3], k] * IL[i, k]
// Block: 256 threads = 8 waves (wave32). Block tile 32(M) x 256(N).
// Wave tile: 16(M) x 64(N) = 4 WMMA 16x16 C-tiles; K=64 -> 16x V_WMMA_F32_16X16X4_F32 each.
__global__ __launch_bounds__(256)
void fpmc_gemm_wmma(const int* __restrict__ X,
                    const float* __restrict__ IL,
                    const float* __restrict__ LI,
                    float* __restrict__ out)
{
    const int lane = threadIdx.x & 31;
    const int wave = threadIdx.x >> 5;
    const int lm   = lane & 15;   // lane within half-wave: M for A, N for B
    const int lh   = lane >> 4;   // half-wave select: K-pair offset

    const int wm = wave >> 2;     // 0..1 -> M sub-tile
    const int wn = wave & 3;      // 0..3 -> N sub-strip

    const int m0     = blockIdx.y * 32 + wm * 16;
    const int n_base = blockIdx.x * 256 + wn * 64;

    // --- A gather: this lane's row index (lanes 0-15 and 16-31 duplicate the gather)
    const int arow = m0 + lm;                 // < 1024 always (grid exact in M)
    const int idx  = X[arow * 40 + 35];       // X[b, 8, 3], flattened (1024,10,4)
    const float* aptr = LI + (size_t)idx * K_IL + 2 * lh;

    // --- B pointers for the 4 N-tiles (clamp rows so OOB tiles load safely;
    //     validity is wave-uniform since N_ITEMS % 16 == 0)
    const int nt0 = n_base, nt1 = n_base + 16, nt2 = n_base + 32, nt3 = n_base + 48;
    const int c0 = (nt0 + lm < N_ITEMS) ? (nt0 + lm) : (N_ITEMS - 1);
    const int c1 = (nt1 + lm < N_ITEMS) ? (nt1 + lm) : (N_ITEMS - 1);
    const int c2 = (nt2 + lm < N_ITEMS) ? (nt2 + lm) : (N_ITEMS - 1);
    const int c3 = (nt3 + lm < N_ITEMS) ? (nt3 + lm) : (N_ITEMS - 1);
    const float* bptr0 = IL + (size_t)c0 * K_IL + 2 * lh;
    const float* bptr1 = IL + (size_t)c1 * K_IL + 2 * lh;
    const float* bptr2 = IL + (size_t)c2 * K_IL + 2 * lh;
    const float* bptr3 = IL + (size_t)c3 * K_IL + 2 * lh;

    v8f acc0 = {}, acc1 = {}, acc2 = {}, acc3 = {};

    #pragma unroll
    for (int k0 = 0; k0 < K_IL; k0 += 4) {
        // A 16x4 frag: lane(lm)=M, v[j] = A[M][k0 + 2*lh + j]
        v2f a  = *(const v2f*)(aptr + k0);
        // B 4x16 frags: lane(lm)=N, v[j] = B[k0 + 2*lh + j][N] = IL[N][k0 + 2*lh + j]
        v2f b0 = *(const v2f*)(bptr0 + k0);
        v2f b1 = *(const v2f*)(bptr1 + k0);
        v2f b2 = *(const v2f*)(bptr2 + k0);
        v2f b3 = *(const v2f*)(bptr3 + k0);

        acc0 = __builtin_amdgcn_wmma_f32_16x16x4_f32(false, a, false, b0, (short)0, acc0, false, false);
        acc1 = __builtin_amdgcn_wmma_f32_16x16x4_f32(false, a, false, b1, (short)0, acc1, false, false);
        acc2 = __builtin_amdgcn_wmma_f32_16x16x4_f32(false, a, false, b2, (short)0, acc2, false, false);
        acc3 = __builtin_amdgcn_wmma_f32_16x16x4_f32(false, a, false, b3, (short)0, acc3, false, false);
    }

    // --- store: C layout: v[r] -> row m0 + r + 8*lh, col nt + lm. NT stores: output
    // is a 410 MB write-once stream; keep it out of L2 so IL stays resident.
    const int rbase = m0 + 8 * lh;
    const int col   = lm;
    #define STORE_TILE(ACC, NT)                                                      \
        if ((NT) < N_ITEMS) {                                                        \
            _Pragma("unroll")                                                        \
            for (int r = 0; r < 8; ++r) {                                            \
                __builtin_nontemporal_store(ACC[r] * 0.125f,                         \
                    out + (size_t)(rbase + r) * N_ITEMS + (NT) + col);               \
            }                                                                        \
        }
    STORE_TILE(acc0, nt0)
    STORE_TILE(acc1, nt1)
    STORE_TILE(acc2, nt2)
    STORE_TILE(acc3, nt3)
    #undef STORE_TILE
}

extern "C" void kernel_launch(void* const* d_in, const int* in_sizes, int n_in,
                              void* d_out, int out_size, void* d_ws, size_t ws_size,
                              hipStream_t stream) {
    (void)in_sizes; (void)n_in; (void)d_ws; (void)ws_size; (void)out_size;
    const int*   X  = (const int*)d_in[0];
    const float* IL = (const float*)d_in[1];
    const float* LI = (const float*)d_in[2];
    float* out = (float*)d_out;

    dim3 block(256, 1, 1);
    dim3 grid((N_ITEMS + 255) / 256, M_ROWS / 32, 1);  // (391, 32)
    fpmc_gemm_wmma<<<grid, block, 0, stream>>>(X, IL, LI, out);
}